// EntMaxBisectDelegator_71296457114412
// MI455X (gfx1250) — compile-verified
//
#include <hip/hip_runtime.h>
#include <hip/hip_bf16.h>
#include <stdint.h>

// ---------------------------------------------------------------------------
// alpha-entmax via 50-step bisection, one workgroup per row (2048 x 32000 f32).
// MI455X plan:
//   * HBM floor = (262MB in + 262MB out)/23.3TB/s ~= 22.5us -> touch each
//     input element exactly once from HBM; the 50-iteration bisection lives
//     entirely in VGPRs (32 fp32 regs/thread, 1024 thr = 32 waves per row).
//   * Stage HBM -> LDS with the CDNA5 Tensor Data Mover (tensor_load_to_lds,
//     TENSORcnt) issued once by wave 0, then LDS -> VGPR via ds_load_b128.
//   * Per bisection step: wave32 shuffle tree -> 32 partials in LDS ->
//     ONE barrier -> every wave redundantly reduces the 32 partials
//     (double-buffered LDS so a single barrier per step is WAR-safe).
//   * alpha=1.5 => exponent 1/(alpha-1)==2 -> z*z FMA fast path (uniform
//     branch); generic path uses native v_log_f32/v_exp_f32.
// ---------------------------------------------------------------------------

#define TPB   1024
#define EPT   32              // elements per thread; TPB*EPT = 32768 >= 32000
#define LDSN  (TPB * EPT)     // staged floats per row (128 KB LDS)

typedef unsigned int v4u __attribute__((ext_vector_type(4)));
typedef int          v8i __attribute__((ext_vector_type(8)));
typedef int          v4i __attribute__((ext_vector_type(4)));

// ---------------- wave32 reductions ----------------------------------------
__device__ __forceinline__ float waveReduceSum(float v) {
#pragma unroll
  for (int off = 16; off > 0; off >>= 1) v += __shfl_xor(v, off, 32);
  return v;
}
__device__ __forceinline__ float waveReduceMax(float v) {
#pragma unroll
  for (int off = 16; off > 0; off >>= 1) v = fmaxf(v, __shfl_xor(v, off, 32));
  return v;
}

// ---------------- single-barrier block reductions --------------------------
// buf: 32 floats. Caller must strictly alternate between two buffers across
// consecutive calls; the intervening call's barrier makes buffer reuse
// WAR-safe. Result is returned in ALL lanes of ALL waves (each wave
// redundantly reduces the 32 partials -> no broadcast round-trip).
__device__ __forceinline__ float blockReduceSum1(float v, float* buf) {
  const int lane = threadIdx.x & 31, wv = threadIdx.x >> 5;
  v = waveReduceSum(v);
  if (lane == 0) buf[wv] = v;
  __syncthreads();
  return waveReduceSum(buf[lane]);   // exactly 32 waves in a 1024 block
}
__device__ __forceinline__ float blockReduceMax1(float v, float* buf) {
  const int lane = threadIdx.x & 31, wv = threadIdx.x >> 5;
  v = waveReduceMax(v);
  if (lane == 0) buf[wv] = v;
  __syncthreads();
  return waveReduceMax(buf[lane]);
}

// ---------------- p(z) = max(z,0)^(1/(alpha-1)) ----------------------------
template <bool SQ>
__device__ __forceinline__ float pfun(float z, float e) {
  z = fmaxf(z, 0.0f);
  if (SQ) return z * z;                              // alpha == 1.5 fast path
  return (z > 0.0f) ? __builtin_exp2f(e * __builtin_log2f(z)) : 0.0f;
}

// ---------------- TDM staging: one row HBM -> LDS --------------------------
__device__ __forceinline__ void tdm_stage_row(const float* __restrict__ rowp,
                                              float* smem, int rowLen) {
#if __has_builtin(__builtin_amdgcn_tensor_load_to_lds)
  if (threadIdx.x < 32) {                    // wave 0 only: one DMA per row
    const uint64_t ga  = (uint64_t)(uintptr_t)rowp;
    const uint32_t lds = (uint32_t)(uintptr_t)smem;  // low 32b = LDS offset
    const uint32_t rl  = (uint32_t)rowLen;
    v4u g0;
    g0[0] = 1u;                                        // count=1, user D#
    g0[1] = lds;                                       // lds_addr
    g0[2] = (uint32_t)ga;                              // global_addr[31:0]
    g0[3] = (uint32_t)((ga >> 32) & 0x1FFFFFFu) | (2u << 30); // [56:32]|type=2
    v8i g1;
    g1[0] = (int)(2u << 16);                 // wg_mask=0, data_size=4B, no pad
    g1[1] = (int)((rl & 0xFFFFu) << 16);     // tensor_dim0[15:0]
    g1[2] = (int)(((rl >> 16) & 0xFFFFu) | (1u << 16)); // dim0 hi | tensor_dim1=1
    g1[3] = (int)((rl & 0xFFFFu) << 16);     // tensor_dim1 hi=0 | tile_dim0=rowLen
    g1[4] = 1;                               // tile_dim1=1, tile_dim2=0
    g1[5] = (int)rl;                         // tensor_dim0_stride[31:0]
    g1[6] = 0;                               // stride hi | dim1_stride lo
    g1[7] = 0;
    v4i z4 = {0, 0, 0, 0};
#if __clang_major__ >= 23
    v8i z8 = {0, 0, 0, 0, 0, 0, 0, 0};
    __builtin_amdgcn_tensor_load_to_lds(g0, g1, z4, z4, z8, 0);
#else
    __builtin_amdgcn_tensor_load_to_lds(g0, g1, z4, z4, 0);
#endif
    __builtin_amdgcn_s_wait_tensorcnt((short)0);
  }
#else
  // Fallback: cooperative b128 copy HBM -> LDS
  const float4* src = reinterpret_cast<const float4*>(rowp);
  float4* dst = reinterpret_cast<float4*>(smem);
  const int n4 = rowLen >> 2;
  for (int i = threadIdx.x; i < n4; i += blockDim.x) dst[i] = src[i];
  for (int i = (n4 << 2) + threadIdx.x; i < rowLen; i += blockDim.x)
    smem[i] = rowp[i];
#endif
}

// ---------------- bisection loop + store (templated on fast path) ----------
// Reduction-buffer parity: caller used red+0 for the max reduce, so here
// f_lo uses red+32, and iteration `it` uses red + 32*(it&1) -> strict
// alternation across all consecutive reduction calls.
template <bool SQ>
__device__ __forceinline__ void run_bisect(const float (&X)[EPT], float tau_lo,
                                           float tau_hi, float e, float* red,
                                           float* __restrict__ outRow, int base,
                                           int rowLen) {
  float s = 0.0f;
#pragma unroll
  for (int j = 0; j < EPT; ++j) s += pfun<SQ>(X[j] - tau_lo, e);
  const float f_lo = blockReduceSum1(s, red + 32) - 1.0f;

  float dm = tau_hi - tau_lo;
  float tlo = tau_lo, tau_m = tau_lo, f_m = f_lo;
#pragma unroll 1
  for (int it = 0; it < 50; ++it) {
    dm *= 0.5f;
    tau_m = tlo + dm;
    float s2 = 0.0f;
#pragma unroll
    for (int j = 0; j < EPT; ++j) s2 += pfun<SQ>(X[j] - tau_m, e);
    f_m = blockReduceSum1(s2, red + 32 * (it & 1)) - 1.0f;
    tlo = (f_m * f_lo >= 0.0f) ? tau_m : tlo;   // uniform per row
  }

  const float inv = 1.0f / (f_m + 1.0f);        // sum(p_m) = f_m + 1
#pragma unroll
  for (int i = 0; i < EPT / 4; ++i) {
    const int idx = base + 4 * i;
    if (idx + 3 < rowLen) {
      float4 o;
      o.x = pfun<SQ>(X[4 * i + 0] - tau_m, e) * inv;
      o.y = pfun<SQ>(X[4 * i + 1] - tau_m, e) * inv;
      o.z = pfun<SQ>(X[4 * i + 2] - tau_m, e) * inv;
      o.w = pfun<SQ>(X[4 * i + 3] - tau_m, e) * inv;
      *reinterpret_cast<float4*>(outRow + idx) = o;   // 16B aligned
    } else {
#pragma unroll
      for (int k = 0; k < 4; ++k)
        if (idx + k < rowLen)
          outRow[idx + k] = pfun<SQ>(X[4 * i + k] - tau_m, e) * inv;
    }
  }
}

// ---------------- kernel ---------------------------------------------------
__global__ __launch_bounds__(TPB) void entmax_bisect_kernel(
    const float* __restrict__ x, const float* __restrict__ alphap,
    float* __restrict__ out, int rowLen) {
  __shared__ float smem[LDSN];   // 128 KB staged row (<= 320 KB/WG on CDNA5)
  __shared__ float red[64];      // two 32-partial buffers (double-buffered)

  const int t = threadIdx.x;
  const int row = blockIdx.x;
  const float alpha = alphap[0];
  const float am1 = alpha - 1.0f;
  const float e = 1.0f / am1;
  const float* rowp = x + (size_t)row * (size_t)rowLen;
  float* outRow = out + (size_t)row * (size_t)rowLen;

  tdm_stage_row(rowp, smem, rowLen);
  __syncthreads();

  // LDS -> registers (8x ds_load_b128 per thread), X = x * (alpha-1)
  float X[EPT];
  const int base = t * EPT;
#pragma unroll
  for (int i = 0; i < EPT / 4; ++i) {
    const float4 a = reinterpret_cast<const float4*>(smem + base)[i];
    X[4 * i + 0] = a.x * am1;
    X[4 * i + 1] = a.y * am1;
    X[4 * i + 2] = a.z * am1;
    X[4 * i + 3] = a.w * am1;
  }
#pragma unroll
  for (int j = 0; j < EPT; ++j)
    if (base + j >= rowLen) X[j] = -INFINITY;   // masked lanes: p=0, max=-inf

  float mx = -INFINITY;
#pragma unroll
  for (int j = 0; j < EPT; ++j) mx = fmaxf(mx, X[j]);
  mx = blockReduceMax1(mx, red + 0);            // parity 0

  const float tau_lo = mx - 1.0f;
  // (1/d)^am1 = exp2(-am1 * log2(d))
  const float tau_hi =
      mx - __builtin_exp2f(-am1 * __builtin_log2f((float)rowLen));

  if (e == 2.0f)    // alpha == 1.5: p(z) = z^2, pure FMA inner loop
    run_bisect<true>(X, tau_lo, tau_hi, e, red, outRow, base, rowLen);
  else
    run_bisect<false>(X, tau_lo, tau_hi, e, red, outRow, base, rowLen);
}

// ---------------- host entry ----------------------------------------------
extern "C" void kernel_launch(void* const* d_in, const int* in_sizes, int n_in,
                              void* d_out, int out_size, void* d_ws,
                              size_t ws_size, hipStream_t stream) {
  (void)n_in; (void)out_size; (void)d_ws; (void)ws_size;
  const float* x = (const float*)d_in[0];
  const float* alpha = (const float*)d_in[1];
  float* out = (float*)d_out;

  const int rowLen = 32000;                 // reference: (2048, 32000)
  const int rows = in_sizes[0] / rowLen;

  entmax_bisect_kernel<<<dim3(rows), dim3(TPB), 0, stream>>>(x, alpha, out,
                                                             rowLen);
}